// HCLDecodeBlock_80427557585266
// MI455X (gfx1250) — compile-verified
//
#include <hip/hip_runtime.h>
#include <hip/hip_bf16.h>
#include <math.h>

#define BATCH   8
#define DMODEL  4096
#define SDIM    16
#define D3      12288
#define FFDIM   16384

typedef float v2f __attribute__((ext_vector_type(2)));
typedef float v8f __attribute__((ext_vector_type(8)));

// WMMA f32 16x16x4 A-operand packed layout:
// element (m, k) -> apack[(k>>2)*64 + ((((k>>1)&1)<<4) + m)*2 + (k&1)]
__device__ __forceinline__ int apack_idx(int m, int k) {
    return ((k >> 2) << 6) + (((((k >> 1) & 1) << 4) + m) << 1) + (k & 1);
}

// ---------------------------------------------------------------------------
// RMSNorm over D=4096 per batch row -> 16-row (8 real + 8 zero-pad) packed A.
// ---------------------------------------------------------------------------
__global__ void rmsnorm_pack_kernel(const float* __restrict__ x,
                                    const float* __restrict__ w,
                                    float* __restrict__ apack) {
    int b = blockIdx.x;
    int tid = threadIdx.x;
    const float* xr = x + b * DMODEL;
    float ss = 0.f;
    for (int k = tid; k < DMODEL; k += 256) { float v = xr[k]; ss += v * v; }
    __shared__ float red[256];
    red[tid] = ss;
    __syncthreads();
    for (int s = 128; s > 0; s >>= 1) {
        if (tid < s) red[tid] += red[tid + s];
        __syncthreads();
    }
    float scale = rsqrtf(red[0] * (1.0f / DMODEL) + 1e-6f);
    for (int k = tid; k < DMODEL; k += 256) {
        float xh = xr[k] * scale * w[k];
        int idx = apack_idx(b, k);
        apack[idx] = xh;        // row m = b
        apack[idx + 16] = 0.f;  // pad row m = b + 8
    }
}

// ---------------------------------------------------------------------------
// Split-K fp32 WMMA GEMM: wave (tile, chunk) accumulates a 16x16 partial tile.
// Weights streamed non-temporally; packed A stays hot in L2.
// partial layout: [chunk][m=0..7][N]
// ---------------------------------------------------------------------------
template <int N, int K, int CH>
__global__ void gemm_splitk_kernel(const float* __restrict__ apack,
                                   const float* __restrict__ Bw,
                                   float* __restrict__ partial) {
    int wave = blockIdx.x * 8 + (threadIdx.x >> 5);
    int tile = wave / CH;
    int c    = wave % CH;
    int n0   = tile << 4;
    constexpr int KC = K / CH;
    int k0 = c * KC;
    int lane = threadIdx.x & 31;
    int l15 = lane & 15;
    int hi  = lane >> 4;
    const float* bcol = Bw + (size_t)(k0 + 2 * hi) * N + n0 + l15;
    const float* ap   = apack + (k0 << 4) + lane * 2;  // (k0/4)*64
    v8f acc = {};
#pragma unroll 8
    for (int kb = 0; kb < (KC >> 2); ++kb) {
        v2f a = *(const v2f*)(ap + (kb << 6));
        v2f bv;
        bv.x = __builtin_nontemporal_load(&bcol[(size_t)(kb << 2) * N]);
        bv.y = __builtin_nontemporal_load(&bcol[(size_t)((kb << 2) + 1) * N]);
        acc = __builtin_amdgcn_wmma_f32_16x16x4_f32(false, a, false, bv,
                                                    (short)0, acc, false, false);
    }
    if (lane < 16) {
#pragma unroll
        for (int i = 0; i < 8; ++i)
            partial[(size_t)(c * 8 + i) * N + n0 + lane] = acc[i];
    }
}

// Dual-B split-K GEMM for the gated MLP (w1 & w3 share the A panel/loop).
template <int N, int K, int CH>
__global__ void gemm_splitk_dual_kernel(const float* __restrict__ apack,
                                        const float* __restrict__ Bw1,
                                        const float* __restrict__ Bw3,
                                        float* __restrict__ p1,
                                        float* __restrict__ p3) {
    int wave = blockIdx.x * 8 + (threadIdx.x >> 5);
    int tile = wave / CH;
    int c    = wave % CH;
    int n0   = tile << 4;
    constexpr int KC = K / CH;
    int k0 = c * KC;
    int lane = threadIdx.x & 31;
    int l15 = lane & 15;
    int hi  = lane >> 4;
    size_t boff = (size_t)(k0 + 2 * hi) * N + n0 + l15;
    const float* b1 = Bw1 + boff;
    const float* b3 = Bw3 + boff;
    const float* ap = apack + (k0 << 4) + lane * 2;
    v8f a1 = {};
    v8f a3 = {};
#pragma unroll 4
    for (int kb = 0; kb < (KC >> 2); ++kb) {
        v2f a = *(const v2f*)(ap + (kb << 6));
        size_t r = (size_t)(kb << 2) * N;
        v2f bv1, bv3;
        bv1.x = __builtin_nontemporal_load(&b1[r]);
        bv1.y = __builtin_nontemporal_load(&b1[r + N]);
        bv3.x = __builtin_nontemporal_load(&b3[r]);
        bv3.y = __builtin_nontemporal_load(&b3[r + N]);
        a1 = __builtin_amdgcn_wmma_f32_16x16x4_f32(false, a, false, bv1,
                                                   (short)0, a1, false, false);
        a3 = __builtin_amdgcn_wmma_f32_16x16x4_f32(false, a, false, bv3,
                                                   (short)0, a3, false, false);
    }
    if (lane < 16) {
#pragma unroll
        for (int i = 0; i < 8; ++i) {
            p1[(size_t)(c * 8 + i) * N + n0 + lane] = a1[i];
            p3[(size_t)(c * 8 + i) * N + n0 + lane] = a3[i];
        }
    }
}

// ---------------------------------------------------------------------------
// Fused: reduce proj split-K partials (u) + FIR tap + gating + diagonal IIR +
// residue contraction; writes new_fir, new_iir outputs and packed y.
// ---------------------------------------------------------------------------
template <int CH>
__global__ void fir_iir_kernel(const float* __restrict__ up,      // [CH][8][D3]
                               const float* __restrict__ fir_state,
                               const float* __restrict__ iir_state,
                               const float* __restrict__ sf_weight,
                               const float* __restrict__ sf_bias,
                               const float* __restrict__ D_res,
                               const float* __restrict__ residues,
                               const float* __restrict__ log_poles,
                               float* __restrict__ out_fir,
                               float* __restrict__ out_iir,
                               float* __restrict__ ypack) {
    int t = blockIdx.x * 256 + threadIdx.x;   // 0..32767
    int b = t >> 12;
    int d = t & 4095;
    int head = d >> 7;
    int i    = d & 127;
    int jbase = head * 384 + i;
    float zp[3];
#pragma unroll
    for (int p = 0; p < 3; ++p) {
        int j  = jbase + (p << 7);
        int fj = b * D3 + j;
        float uv = 0.f;
#pragma unroll
        for (int c = 0; c < CH; ++c) uv += up[(size_t)(c * 8 + b) * D3 + j];
        float f0 = fir_state[2 * fj + 0];
        float f1 = fir_state[2 * fj + 1];
        const float* wj = sf_weight + j * 3;
        zp[p] = wj[2] * uv + f0 * wj[0] + f1 * wj[1] + sf_bias[j];
        out_fir[2 * fj + 0] = f1;
        out_fir[2 * fj + 1] = uv;
    }
    float x2 = zp[0], x1 = zp[1], vv = zp[2];
    float x1v = x1 * vv;
    float acc = 0.f;
    int base = (b * DMODEL + d) * SDIM;
    int rb   = d * SDIM;
#pragma unroll
    for (int s = 0; s < SDIM; ++s) {
        float pole = expf(log_poles[rb + s]);
        float ni   = pole * iir_state[base + s] + x1v;
        out_iir[base + s] = ni;
        acc += residues[rb + s] * ni;
    }
    float y = x2 * (acc + D_res[d] * x1v);
    int idx = apack_idx(b, d);
    ypack[idx] = y;
    ypack[idx + 16] = 0.f;
}

// Reduce out-proj partials: xmid = sum_c p + out_b + x
template <int CH>
__global__ void reduce_out_kernel(const float* __restrict__ p,
                                  const float* __restrict__ out_b,
                                  const float* __restrict__ x,
                                  float* __restrict__ xmid) {
    int t = blockIdx.x * 256 + threadIdx.x;   // 0..32767
    int m = t >> 12;
    int n = t & 4095;
    float s = out_b[n] + x[m * DMODEL + n];
#pragma unroll
    for (int c = 0; c < CH; ++c) s += p[(size_t)(c * 8 + m) * DMODEL + n];
    xmid[m * DMODEL + n] = s;
}

// Reduce MLP13 partials, apply SiLU gate, scatter into packed layout (+pad 0)
template <int CH>
__global__ void reduce_mlp13_kernel(const float* __restrict__ p1,
                                    const float* __restrict__ p3,
                                    float* __restrict__ gpack) {
    int t = blockIdx.x * 256 + threadIdx.x;   // 0..131071
    int m = t >> 14;
    int n = t & 16383;
    float h = 0.f, g3 = 0.f;
#pragma unroll
    for (int c = 0; c < CH; ++c) {
        size_t o = (size_t)(c * 8 + m) * FFDIM + n;
        h  += p1[o];
        g3 += p3[o];
    }
    float g = (h / (1.f + expf(-h))) * g3;
    int idx = apack_idx(m, n);
    gpack[idx] = g;
    gpack[idx + 16] = 0.f;   // pad row m + 8
}

// Reduce MLP2 partials: x_final = xmid + sum_c p
template <int CH>
__global__ void reduce_mlp2_kernel(const float* __restrict__ p,
                                   const float* __restrict__ xmid,
                                   float* __restrict__ xout) {
    int t = blockIdx.x * 256 + threadIdx.x;   // 0..32767
    int m = t >> 12;
    int n = t & 4095;
    float s = xmid[m * DMODEL + n];
#pragma unroll
    for (int c = 0; c < CH; ++c) s += p[(size_t)(c * 8 + m) * DMODEL + n];
    xout[m * DMODEL + n] = s;
}

extern "C" void kernel_launch(void* const* d_in, const int* in_sizes, int n_in,
                              void* d_out, int out_size, void* d_ws, size_t ws_size,
                              hipStream_t stream) {
    (void)in_sizes; (void)n_in; (void)out_size; (void)ws_size;
    const float* x           = (const float*)d_in[0];
    const float* fir_state   = (const float*)d_in[1];
    const float* iir_state   = (const float*)d_in[2];
    const float* pre_norm_w  = (const float*)d_in[3];
    const float* proj_w      = (const float*)d_in[4];
    const float* sf_weight   = (const float*)d_in[5];
    const float* sf_bias     = (const float*)d_in[6];
    const float* D_res       = (const float*)d_in[7];
    const float* residues    = (const float*)d_in[8];
    const float* log_poles   = (const float*)d_in[9];
    const float* out_w       = (const float*)d_in[10];
    const float* out_b       = (const float*)d_in[11];
    const float* post_norm_w = (const float*)d_in[12];
    const float* mlp_w1      = (const float*)d_in[13];
    const float* mlp_w3      = (const float*)d_in[14];
    const float* mlp_w2      = (const float*)d_in[15];

    float* out_x   = (float*)d_out;            // (8, 1, 4096)
    float* out_fir = out_x + BATCH * DMODEL;   // (8, 12288, 2)
    float* out_iir = out_fir + BATCH * D3 * 2; // (8, 4096, 16)

    float* ws     = (float*)d_ws;
    float* apack  = ws;                        // 16*4096   = 65536
    float* ypack  = apack + 16 * DMODEL;       // 16*4096   = 65536
    float* xmid   = ypack + 16 * DMODEL;       // 8*4096    = 32768
    float* xnpack = xmid + BATCH * DMODEL;     // 16*4096   = 65536
    float* gpack  = xnpack + 16 * DMODEL;      // 16*16384  = 262144
    float* pred   = gpack + 16 * FFDIM;        // shared split-K partial region
    // pred usage (sequential stages, region reused):
    //  proj : 4 chunks * 8 * 12288           =  393216
    //  out  : 8 chunks * 8 * 4096            =  262144
    //  mlp13: 2 * (4 chunks * 8 * 16384)     = 1048576  (max)
    //  mlp2 : 8 chunks * 8 * 4096            =  262144
    float* p1 = pred;
    float* p3 = pred + 4 * 8 * FFDIM;

    // 1) pre-RMSNorm -> packed A
    rmsnorm_pack_kernel<<<BATCH, 256, 0, stream>>>(x, pre_norm_w, apack);
    // 2) proj GEMM (16x4096)x(4096x12288), split-K=4: 3072 waves
    gemm_splitk_kernel<D3, DMODEL, 4>
        <<<(D3 / 16) * 4 / 8, 256, 0, stream>>>(apack, proj_w, pred);
    // 3) fused u-reduce + FIR + gating + IIR -> outputs + packed y
    fir_iir_kernel<4><<<BATCH * DMODEL / 256, 256, 0, stream>>>(
        pred, fir_state, iir_state, sf_weight, sf_bias, D_res, residues,
        log_poles, out_fir, out_iir, ypack);
    // 4) out-proj GEMM (16x4096)x(4096x4096), split-K=8: 2048 waves
    gemm_splitk_kernel<DMODEL, DMODEL, 8>
        <<<(DMODEL / 16) * 8 / 8, 256, 0, stream>>>(ypack, out_w, pred);
    reduce_out_kernel<8><<<BATCH * DMODEL / 256, 256, 0, stream>>>(
        pred, out_b, x, xmid);
    // 5) post-RMSNorm -> packed A
    rmsnorm_pack_kernel<<<BATCH, 256, 0, stream>>>(xmid, post_norm_w, xnpack);
    // 6) gated MLP up: dual GEMM (16x4096)x(4096x16384), split-K=4: 4096 waves
    gemm_splitk_dual_kernel<FFDIM, DMODEL, 4>
        <<<(FFDIM / 16) * 4 / 8, 256, 0, stream>>>(xnpack, mlp_w1, mlp_w3, p1, p3);
    reduce_mlp13_kernel<4><<<BATCH * FFDIM / 256, 256, 0, stream>>>(p1, p3, gpack);
    // 7) MLP down: (16x16384)x(16384x4096), split-K=8: 2048 waves
    gemm_splitk_kernel<DMODEL, FFDIM, 8>
        <<<(DMODEL / 16) * 8 / 8, 256, 0, stream>>>(gpack, mlp_w2, pred);
    reduce_mlp2_kernel<8><<<BATCH * DMODEL / 256, 256, 0, stream>>>(
        pred, xmid, out_x);
}